// transformationMatMulti_7653631721982
// MI455X (gfx1250) — compile-verified
//
#include <hip/hip_runtime.h>

typedef __attribute__((ext_vector_type(2))) float v2f;
typedef __attribute__((ext_vector_type(4))) float v4f;
typedef __attribute__((ext_vector_type(8))) float v8f;

// Problem constants (from reference): B=4096, N_IN=64, N_OUT=64
#define NB 4096
#define NJ 64
#define NK 64

// ---------------------------------------------------------------------------
// Phase 1: build the 4096 4x4 transform matrices WM[j][k] and store them
// pre-swizzled into the WMMA B-operand lane layout:
//   WMx[((j*16 + kg)*32 + L)*2 + v] = WM[j][4*kg + ((L>>2)&3)][2*(L>>4)+v][L&3]
// so that kernel 2 can fetch its B operand with one global_load_b64 per lane.
// Size: 64*16*32*2 floats = 256 KB (lives in d_ws).
// ---------------------------------------------------------------------------
__global__ __launch_bounds__(256) void build_wm_kernel(
    const float* __restrict__ aW, const float* __restrict__ uW,
    const float* __restrict__ tW, float* __restrict__ WMx) {
  int m = blockIdx.x * blockDim.x + threadIdx.x;
  if (m >= NJ * NK) return;

  float a  = aW[m];
  float s0 = 1.0f / (1.0f + expf(-uW[3 * m + 0]));
  float s1 = 1.0f / (1.0f + expf(-uW[3 * m + 1]));
  float s2 = 1.0f / (1.0f + expf(-uW[3 * m + 2]));
  float inv = rsqrtf(s0 * s0 + s1 * s1 + s2 * s2);
  float ux = s0 * inv, uy = s1 * inv, uz = s2 * inv;
  float sa = sinf(a), ca = cosf(a);

  // rot = sin(a)*K + cos(a)*(I3 - u u^T) + u u^T ; K = [[0,-uz,uy],[uz,0,-ux],[-uy,ux,0]]
  float W[4][4];
  W[0][0] = ca * (1.0f - ux * ux) + ux * ux;
  W[0][1] = -sa * uz - ca * ux * uy + ux * uy;
  W[0][2] =  sa * uy - ca * ux * uz + ux * uz;
  W[1][0] =  sa * uz - ca * uy * ux + uy * ux;
  W[1][1] = ca * (1.0f - uy * uy) + uy * uy;
  W[1][2] = -sa * ux - ca * uy * uz + uy * uz;
  W[2][0] = -sa * uy - ca * uz * ux + uz * ux;
  W[2][1] =  sa * ux - ca * uz * uy + uz * uy;
  W[2][2] = ca * (1.0f - uz * uz) + uz * uz;
  W[0][3] = tW[3 * m + 0];
  W[1][3] = tW[3 * m + 1];
  W[2][3] = tW[3 * m + 2];
  W[3][0] = 0.0f; W[3][1] = 0.0f; W[3][2] = 0.0f; W[3][3] = 1.0f;

  int j = m >> 6, k = m & 63;
  int kg = k >> 2, ci = k & 3;
#pragma unroll
  for (int h = 0; h < 2; ++h) {
#pragma unroll
    for (int z = 0; z < 4; ++z) {
      int L = 16 * h + 4 * ci + z;
      v2f val = {W[2 * h][z], W[2 * h + 1][z]};
      *(v2f*)(WMx + ((size_t)(j * 16 + kg) * 32 + L) * 2) = val;
    }
  }
}

// ---------------------------------------------------------------------------
// Phase 2: each wave owns a 4-batch x 4-k tile (16 chains of 4x4 matrices).
// Per j step:
//   WMMA#1 (dense, 100%): P(16x16) = [I[b0..b3,j] stacked](16x4) @ [WM[j,k0..k3] side](4x16)
//     -> P block (bi,kj) = A_{bi,kj} = I[b,j] @ WM[j,k] for all 16 chains.
//   P -> LDS (lane-major), then 4 cross-packed WMMAs:
//   WMMA#2 group g: D = [carry(g,0..3) stacked](16x4) @ [A(g,0..3) side](4x16);
//     diagonal 4x4 blocks are the updated carries, written back to LDS.
// Carries init to A at j=0 (chain = A_0 @ A_1 @ ... @ A_63).
// ---------------------------------------------------------------------------
__global__ __launch_bounds__(256) void chain_wmma_kernel(
    const float* __restrict__ I, const float* __restrict__ WMx,
    float* __restrict__ O) {
  __shared__ float lds[8][512];  // per-wave: 256 floats Pbuf + 256 floats carries

  const int tid = threadIdx.x;
  const int w   = tid >> 5;
  const int L   = tid & 31;

  const int wave_global = blockIdx.x * 8 + w;
  const int bt = wave_global >> 4;   // 0..1023, shared by all 8 waves of a block
  const int kg = wave_global & 15;   // 0..15
  const int b0 = bt * 4;

  float* Pbuf = &lds[w][0];    // Pbuf[l*8 + r] = P element (M = r + 8*(l>=16), N = l&15)
  float* Cc   = &lds[w][256];  // Cc[chain*16 + y*4 + x] = carry[x][y] (chain = g*4+ci)

  const int h  = L >> 4;       // lane half
  const int Ll = L & 15;
  const int ci = Ll >> 2;      // sub-block index within a 16-wide dimension
  const int xz = Ll & 3;       // element index within a 4-block
  const int y0 = 2 * h;        // K base for this lane half (A/B operand layout)

  // A operand (Istack): lane -> I[b0+ci, j, x=xz, y0..y0+1]
  const float* Iptr = I + ((size_t)(b0 + ci) * NJ) * 16 + xz * 4 + y0;
  // B operand (WM, pre-swizzled): one b64 per lane per j
  const float* Wptr = WMx + ((size_t)kg * 32 + L) * 2;

  const v8f zc = {};  // zero accumulator

  for (int j = 0; j < NJ; ++j) {
    v2f Ist = *(const v2f*)(Iptr + j * 16);
    v2f Wms = *(const v2f*)(Wptr + (size_t)j * 1024);

    // Dense WMMA: all 16 per-step matrices A_{bi,kj} in one shot.
    v8f P = __builtin_amdgcn_wmma_f32_16x16x4_f32(false, Ist, false, Wms,
                                                  (short)0, zc, false, false);

    if (j == 0) {
      // carries := A_0 ; lane holds columns (kj=ci, z=xz) of A for bi=2h and 2h+1
      v4f lo = {P[0], P[1], P[2], P[3]};
      v4f hi = {P[4], P[5], P[6], P[7]};
      *(v4f*)(Cc + ((2 * h) * 4 + ci) * 16 + xz * 4)     = lo;  // chain (bi=2h,   kj=ci)
      *(v4f*)(Cc + ((2 * h + 1) * 4 + ci) * 16 + xz * 4) = hi;  // chain (bi=2h+1, kj=ci)
    } else {
      // Stage P through LDS so groups can fetch it in B-operand layout.
      *(v4f*)(Pbuf + L * 8)     = (v4f){P[0], P[1], P[2], P[3]};
      *(v4f*)(Pbuf + L * 8 + 4) = (v4f){P[4], P[5], P[6], P[7]};

#pragma unroll
      for (int g = 0; g < 4; ++g) {
        // Cstack (A operand): carry_{g,ci}[x=xz][y0..y0+1]  (Cc stores [y][x])
        v2f Cs;
        Cs[0] = Cc[(g * 4 + ci) * 16 + (y0 + 0) * 4 + xz];
        Cs[1] = Cc[(g * 4 + ci) * 16 + (y0 + 1) * 4 + xz];
        // B operand: A_{g,cj=ci}[y0..y0+1][z=xz] gathered from Pbuf
        v2f Bo = *(const v2f*)(Pbuf + (size_t)(((g >= 2) ? 16 : 0) + Ll) * 8 +
                               4 * (g & 1) + y0);

        v8f D = __builtin_amdgcn_wmma_f32_16x16x4_f32(false, Cs, false, Bo,
                                                      (short)0, zc, false, false);

        // Diagonal block (ci,ci) -> new carry; present in this lane iff h == ci>>1.
        if (h == (ci >> 1)) {
          v4f dv = (ci & 1) ? (v4f){D[4], D[5], D[6], D[7]}
                            : (v4f){D[0], D[1], D[2], D[3]};
          *(v4f*)(Cc + (g * 4 + ci) * 16 + xz * 4) = dv;  // [y=xz][x=0..3]
        }
      }
    }
  }

  // Epilogue: O[b,k,x,y] = carry[x][y] = Cc[chain*16 + y*4 + x]
  const int ch = L >> 1;  // chain 0..15
  const int hf = L & 1;   // which 8 of the 16 output elements
  float o[8];
#pragma unroll
  for (int t = 0; t < 8; ++t) {
    int idx = hf * 8 + t;
    int x = idx >> 2, y = idx & 3;
    o[t] = Cc[ch * 16 + y * 4 + x];
  }
  const int b = b0 + (ch >> 2);
  const int k = kg * 4 + (ch & 3);
  float* Op = O + ((size_t)b * NK + k) * 16 + hf * 8;
  *(v4f*)(Op)     = (v4f){o[0], o[1], o[2], o[3]};
  *(v4f*)(Op + 4) = (v4f){o[4], o[5], o[6], o[7]};
}

// ---------------------------------------------------------------------------
extern "C" void kernel_launch(void* const* d_in, const int* in_sizes, int n_in,
                              void* d_out, int out_size, void* d_ws, size_t ws_size,
                              hipStream_t stream) {
  const float* I  = (const float*)d_in[0];  // (4096, 64, 4, 4)
  const float* aW = (const float*)d_in[1];  // (4096, 1, 1)
  const float* uW = (const float*)d_in[2];  // (4096, 1, 3)
  const float* tW = (const float*)d_in[3];  // (4096, 1, 3)
  float* O   = (float*)d_out;               // (4096, 64, 4, 4)
  float* WMx = (float*)d_ws;                // 256 KB swizzled weights

  // Phase 1: 4096 transform matrices -> pre-swizzled B-operand layout.
  build_wm_kernel<<<(NJ * NK + 255) / 256, 256, 0, stream>>>(aW, uW, tW, WMx);

  // Phase 2: 1024 b-tiles x 16 k-groups = 16384 waves = 2048 blocks of 8 waves.
  chain_wmma_kernel<<<2048, 256, 0, stream>>>(I, WMx, O);
}